// DistanceMapPenalizingLoss_40458591928838
// MI455X (gfx1250) — compile-verified
//
#include <hip/hip_runtime.h>
#include <stdint.h>

typedef __attribute__((ext_vector_type(2))) float v2f;
typedef __attribute__((ext_vector_type(8))) float v8f;
typedef __attribute__((ext_vector_type(4))) unsigned int v4u;
typedef __attribute__((ext_vector_type(4))) int v4i;
typedef __attribute__((ext_vector_type(8))) int v8i;

#define EDT_B 4
#define EDT_H 320
#define EDT_W 320
#define EDT_ROWS (EDT_B * EDT_H)          // 1280 rows of the [B*H, W] view
#define EDT_N (EDT_B * EDT_H * EDT_W)     // 409600
#define LAMBDA_F 1.0f

#if defined(__gfx1250__) && __has_builtin(__builtin_amdgcn_tensor_load_to_lds) && __has_builtin(__builtin_amdgcn_s_wait_tensorcnt)
#define HAVE_TDM 1
#else
#define HAVE_TDM 0
#endif

// ---------------------------------------------------------------------------
// Kernel A: per-column 1D EDT along H.  g2[b,h,w] = min(d_up, d_dn)^2.
// One thread per (b,w) column; lanes map to consecutive w => coalesced.
// ---------------------------------------------------------------------------
__global__ void edt_cols_kernel(const int* __restrict__ targets,
                                float* __restrict__ g2) {
  int t = blockIdx.x * blockDim.x + threadIdx.x;
  if (t >= EDT_B * EDT_W) return;
  int b = t / EDT_W;
  int w = t - b * EDT_W;
  int base = b * (EDT_H * EDT_W) + w;

  const float BIG = 1.0e9f;
  float d = BIG;
  for (int h = 0; h < EDT_H; ++h) {
    int idx = base + h * EDT_W;
    bool seed = targets[idx] > 0;
    d = seed ? 0.0f : (d + 1.0f);
    g2[idx] = d;                           // stash d_up
  }
  float dn = BIG;
  for (int h = EDT_H - 1; h >= 0; --h) {
    int idx = base + h * EDT_W;
    bool seed = targets[idx] > 0;
    dn = seed ? 0.0f : (dn + 1.0f);
    float m = fminf(g2[idx], dn);
    g2[idx] = m * m;                       // finalize g^2
  }
}

// ---------------------------------------------------------------------------
// TDM: DMA a 320 x f32 contiguous row from global into LDS at lds_byte_off.
// D# group0: count=1 (valid), lds_addr, global_addr(57b), type=2 ("image").
// D# group1: data_size=4B (enc 2), tensor_dim0=tile_dim0=320, tensor_dim1=
//            tile_dim1=1, tensor_dim0_stride=320.  Groups 2/3 zero (<=2D).
// ---------------------------------------------------------------------------
#if HAVE_TDM
__device__ __forceinline__ void tdm_load_row320_f32(const float* gptr,
                                                    unsigned lds_byte_off) {
  uint64_t ga = (uint64_t)(uintptr_t)gptr;
  v4u g0 = { 1u,
             lds_byte_off,
             (unsigned)(ga & 0xFFFFFFFFu),
             (unsigned)((ga >> 32) & 0x01FFFFFFu) | (2u << 30) };
  v8i g1 = { (int)(2u << 16),                   // data_size = 4B
             (int)((unsigned)EDT_W << 16),      // tensor_dim0[15:0] << 16
             (int)(1u << 16),                   // tensor_dim0 hi=0 | tensor_dim1 lo=1
             (int)((unsigned)EDT_W << 16),      // tensor_dim1 hi=0 | tile_dim0=320
             (int)1,                            // tile_dim1=1, tile_dim2=0
             (int)EDT_W,                        // tensor_dim0_stride lo32
             0, 0 };
  v4i gg2 = {0, 0, 0, 0};
  v4i gg3 = {0, 0, 0, 0};
#if __clang_major__ >= 23
  v8i gg4 = {0, 0, 0, 0, 0, 0, 0, 0};
  __builtin_amdgcn_tensor_load_to_lds(g0, g1, gg2, gg3, gg4, 0);
#else
  __builtin_amdgcn_tensor_load_to_lds(g0, g1, gg2, gg3, 0);
#endif
}
#endif

// ---------------------------------------------------------------------------
// Wave-level exact 64-per-instruction f32 sum using V_WMMA_F32_16X16X4_F32
// with B == all-ones (layout-agnostic).  Result valid on lane 0.
// ---------------------------------------------------------------------------
__device__ __forceinline__ float wmma_sum_reduce(const float* vals, int nchunks, int lane) {
  v8f c = {0.f, 0.f, 0.f, 0.f, 0.f, 0.f, 0.f, 0.f};
  v2f ones; ones.x = 1.0f; ones.y = 1.0f;
  for (int ch = 0; ch < nchunks; ++ch) {
    v2f a;
    a.x = vals[ch * 64 + lane * 2 + 0];
    a.y = vals[ch * 64 + lane * 2 + 1];
    // D[m,n] = sum_k A[m,k] * 1 + C[m,n]  -> every column holds row-sums
    c = __builtin_amdgcn_wmma_f32_16x16x4_f32(false, a, false, ones,
                                              (short)0, c, false, false);
  }
  float s = c[0] + c[1] + c[2] + c[3] + c[4] + c[5] + c[6] + c[7];
  // lanes 0-15 hold sum of rows 0..7 (per column), lanes 16-31 rows 8..15
  return s + __shfl_xor(s, 16, 32);
}

// ---------------------------------------------------------------------------
// Kernel B: per-row min-plus transform + fused sigmoid/sqrt/weighting and a
// deterministic per-row WMMA reduction.  One block (320 threads) per row.
// Both the g2 row and the logits row are staged into LDS by the TDM.
// ---------------------------------------------------------------------------
__global__ __launch_bounds__(EDT_W) void edt_rows_loss_kernel(
    const float* __restrict__ logits,
    const float* __restrict__ g2,
    float* __restrict__ partials) {
  __shared__ __align__(16) float smem[3 * EDT_W];
  float* sg2  = smem;                      // [0, 320)
  float* slog = smem + EDT_W;              // [320, 640)
  float* sred = smem + 2 * EDT_W;          // [640, 960)

  const int r = blockIdx.x;                // row in [B*H, W] view
  const int tid = threadIdx.x;             // 0..319
  const float* g2row  = g2     + (size_t)r * EDT_W;
  const float* logrow = logits + (size_t)r * EDT_W;

#if HAVE_TDM
  if (tid < 32) {                          // wave 0 issues both DMAs
    tdm_load_row320_f32(g2row, 0u);
    tdm_load_row320_f32(logrow, (unsigned)(EDT_W * sizeof(float)));
    __builtin_amdgcn_s_wait_tensorcnt(0);  // both TDM ops complete
  }
  __syncthreads();
#else
  sg2[tid]  = g2row[tid];
  slog[tid] = logrow[tid];
  __syncthreads();
#endif

  // min-plus: D2[wo] = min_wi g2[wi] + (wo-wi)^2
  // ds_load_b128 per 4 columns; two independent min-chains for ILP.
  const float wo = (float)tid;
  float d2a = 3.4e37f, d2b = 3.4e37f;
#pragma unroll 4
  for (int wi = 0; wi < EDT_W; wi += 4) {
    float4 gq = *(const float4*)(sg2 + wi);
    float o0 = wo - (float)wi;
    float o1 = o0 - 1.0f;
    float o2 = o0 - 2.0f;
    float o3 = o0 - 3.0f;
    d2a = fminf(d2a, fmaf(o0, o0, gq.x));
    d2b = fminf(d2b, fmaf(o1, o1, gq.y));
    d2a = fminf(d2a, fmaf(o2, o2, gq.z));
    d2b = fminf(d2b, fmaf(o3, o3, gq.w));
  }
  float dist = sqrtf(fminf(d2a, d2b));
  float x = slog[tid];
  float prob = 1.0f / (1.0f + __expf(-x));
  // loss = LAMBDA + (1/N) * sum prob*(dist - LAMBDA)
  float part = prob * (dist - LAMBDA_F);

  sred[tid] = part;
  __syncthreads();

  if (tid < 32) {                          // wave 0, EXEC all ones
    float tot = wmma_sum_reduce(sred, EDT_W / 64, tid);
    if (tid == 0) partials[r] = tot;
  }
}

// ---------------------------------------------------------------------------
// Kernel C: deterministic final reduction of 1280 row partials via WMMA,
// single wave.  out = LAMBDA + total / N.
// ---------------------------------------------------------------------------
__global__ void final_reduce_kernel(const float* __restrict__ partials,
                                    float* __restrict__ out) {
  __shared__ __align__(16) float sp[EDT_ROWS];
  int lane = threadIdx.x;                  // 32 threads
  for (int i = lane; i < EDT_ROWS; i += 32) sp[i] = partials[i];
  __syncthreads();
  float tot = wmma_sum_reduce(sp, EDT_ROWS / 64, lane);
  if (lane == 0) out[0] = LAMBDA_F + tot * (1.0f / (float)EDT_N);
}

// ---------------------------------------------------------------------------
extern "C" void kernel_launch(void* const* d_in, const int* in_sizes, int n_in,
                              void* d_out, int out_size, void* d_ws, size_t ws_size,
                              hipStream_t stream) {
  (void)in_sizes; (void)n_in; (void)out_size; (void)ws_size;
  const float* logits = (const float*)d_in[0];
  const int* targets = (const int*)d_in[1];
  float* out = (float*)d_out;

  float* g2 = (float*)d_ws;                // N floats
  float* partials = g2 + EDT_N;            // EDT_ROWS floats

  edt_cols_kernel<<<(EDT_B * EDT_W + 255) / 256, 256, 0, stream>>>(targets, g2);
  edt_rows_loss_kernel<<<EDT_ROWS, EDT_W, 0, stream>>>(logits, g2, partials);
  final_reduce_kernel<<<1, 32, 0, stream>>>(partials, out);
}